// MultiheadAttention_70068096467570
// MI455X (gfx1250) — compile-verified
//
#include <hip/hip_runtime.h>
#include <math.h>

// MI455X / gfx1250: wave32, WMMA f32 16x16x4, TDM async tensor loads.
typedef __attribute__((ext_vector_type(2))) float v2f;
typedef __attribute__((ext_vector_type(4))) float v4f;
typedef __attribute__((ext_vector_type(8))) float v8f;

#define B_   2
#define T_   2048
#define D_   1024
#define H_   16
#define HD_  64
#define INV_TEMP 0.03125f   // 1/sqrt(1024)
#define LDS_STRIDE 36       // 64-row W slab, padded: conflict-free + 16B aligned

#if defined(__has_builtin)
#if __has_builtin(__builtin_amdgcn_tensor_load_to_lds)
#define HAVE_TDM 1
#endif
#endif

#if HAVE_TDM
typedef unsigned int u32x4 __attribute__((ext_vector_type(4)));
typedef int i32x4 __attribute__((ext_vector_type(4)));
typedef int i32x8 __attribute__((ext_vector_type(8)));

// TENSOR_LOAD_TO_LDS of one W k-slab: tile 32 (k) x 64 (rows), fp32,
// row stride = K elements, LDS padding 4 dwords per 32 dwords -> stride 36.
static __device__ __forceinline__ void tdm_load_slab(const float* gtile,
                                                     unsigned lds_off,
                                                     int K, int N) {
  const unsigned long long ga = (unsigned long long)(uintptr_t)gtile;
  u32x4 g0;
  g0.x = 1u;                                   // count=1 (valid user D#)
  g0.y = lds_off;                              // lds_addr (bytes)
  g0.z = (unsigned)(ga & 0xFFFFFFFFu);         // global_addr[31:0]
  g0.w = (unsigned)((ga >> 32) & 0x01FFFFFFu)  // global_addr[56:32]
         | (2u << 30);                         // type = 2 ("image")
  i32x8 g1;
  g1[0] = (int)((2u << 16)      // data_size = 4 bytes
                | (1u << 20)    // pad_enable
                | (4u << 22)    // pad_interval: code 4 = every 32 dwords
                | (3u << 25));  // pad_amount:   code 3 = 4 dwords
  g1[1] = (int)(((unsigned)K & 0xFFFFu) << 16);            // tensor_dim0 lo16
  g1[2] = (int)((((unsigned)K >> 16) & 0xFFFFu)            // tensor_dim0 hi16
                | (((unsigned)N & 0xFFFFu) << 16));        // tensor_dim1 lo16
  g1[3] = (int)((((unsigned)N >> 16) & 0xFFFFu)            // tensor_dim1 hi16
                | (32u << 16));                            // tile_dim0 = 32
  g1[4] = 64;                                              // tile_dim1 = 64
  g1[5] = K;                                               // dim0_stride lo32
  g1[6] = 0;
  g1[7] = 0;
  const i32x4 z4 = (i32x4)0;
#if defined(__clang_major__) && __clang_major__ >= 23
  __builtin_amdgcn_tensor_load_to_lds(g0, g1, z4, z4, (i32x8)0, 0);
#else
  __builtin_amdgcn_tensor_load_to_lds(g0, g1, z4, z4, 0);
#endif
}
#endif  // HAVE_TDM

static __device__ __forceinline__ float half_reduce_max(float v) {
#pragma unroll
  for (int m = 1; m <= 8; m <<= 1) v = fmaxf(v, __shfl_xor(v, m, 32));
  return v;
}
static __device__ __forceinline__ float half_reduce_sum(float v) {
#pragma unroll
  for (int m = 1; m <= 8; m <<= 1) v += __shfl_xor(v, m, 32);
  return v;
}

static __device__ __forceinline__ v8f wmma_f32(v2f a, v2f b, v8f c) {
  return __builtin_amdgcn_wmma_f32_16x16x4_f32(false, a, false, b, (short)0, c,
                                               false, false);
}

// Y[M,N] = X[M,K] @ W[N,K]^T  (torch Linear, no bias)
// block = 256 threads = 8 waves; block tile 256x64; wave tile 32x64.
// W k-slabs (64 rows x 32 k) double-buffered in LDS; slab kb+1 is DMA'd by
// the TDM (or staged via registers) while the 64 WMMAs of slab kb execute.
__global__ __launch_bounds__(256) void gemm_xwt(const float* __restrict__ X,
                                                const float* __restrict__ W,
                                                float* __restrict__ Y,
                                                int M, int N, int K) {
  __shared__ __align__(16) float Wl[2][64 * LDS_STRIDE];

  const int lane = threadIdx.x & 31;
  const int wv   = threadIdx.x >> 5;
  const int lm   = lane & 15;   // M (A/C) or N (B/C) index
  const int hf   = lane >> 4;   // selects K pair {0,1} vs {2,3}
  const int m0 = blockIdx.y * 256 + wv * 32;
  const int n0 = blockIdx.x * 64;

  const float* Xr0 = X + (size_t)(m0 + lm) * K + 2 * hf;       // subtile 0
  const float* Xr1 = X + (size_t)(m0 + 16 + lm) * K + 2 * hf;  // subtile 1
  const float* Wt  = W + (size_t)n0 * K;                       // block's W rows

  v8f acc[2][4];
#pragma unroll
  for (int mi = 0; mi < 2; ++mi)
#pragma unroll
    for (int nt = 0; nt < 4; ++nt) acc[mi][nt] = (v8f){};

  const int nKB = K / 32;

#if HAVE_TDM
  // prologue: DMA slab 0, wait, publish
  if (wv == 0) {
    tdm_load_slab(Wt, (unsigned)(uintptr_t)&Wl[0][0], K, N);
    __builtin_amdgcn_s_wait_tensorcnt(0);
  }
  __syncthreads();
#else
  // staging assignment: thread -> 8 consecutive floats of one W row
  const int sn = threadIdx.x >> 2;        // 0..63  (W row within slab)
  const int sc = (threadIdx.x & 3) * 8;   // 0,8,16,24
  const float* wsrc = Wt + (size_t)sn * K + sc;
  {
    v4f p0 = *(const v4f*)(wsrc);
    v4f p1 = *(const v4f*)(wsrc + 4);
    float* wdst = &Wl[0][sn * LDS_STRIDE + sc];
    *(v4f*)(wdst)     = p0;
    *(v4f*)(wdst + 4) = p1;
  }
  __syncthreads();
#endif

  for (int kb = 0; kb < nKB; ++kb) {
    const int kbase = kb * 32;
    const float* Wb = &Wl[kb & 1][0];
    const bool more = (kb + 1 < nKB);

    // kick off slab kb+1 (everyone passed the previous barrier, so the
    // destination buffer is no longer being read)
#if HAVE_TDM
    if (wv == 0 && more)
      tdm_load_slab(Wt + kbase + 32,
                    (unsigned)(uintptr_t)&Wl[(kb + 1) & 1][0], K, N);
#else
    v4f st0, st1;
    if (more) {
      st0 = *(const v4f*)(wsrc + kbase + 32);
      st1 = *(const v4f*)(wsrc + kbase + 36);
    }
#endif
    __builtin_prefetch(Xr0 + kbase + 32, 0, 0);  // global_prefetch_b8
    __builtin_prefetch(Xr1 + kbase + 32, 0, 0);

    // batch-load all A fragments for this slab (16 x global_load_b64)
    v2f a0[8], a1[8];
#pragma unroll
    for (int st = 0; st < 8; ++st) {
      a0[st] = *(const v2f*)(Xr0 + kbase + 4 * st);
      a1[st] = *(const v2f*)(Xr1 + kbase + 4 * st);
    }

    // double-buffered B fragments from LDS: load st+1 before WMMAs of st
    const float* wb0 = Wb + lm * LDS_STRIDE + 2 * hf;
    const float* wb1 = wb0 + 16 * LDS_STRIDE;
    const float* wb2 = wb0 + 32 * LDS_STRIDE;
    const float* wb3 = wb0 + 48 * LDS_STRIDE;
    v2f be[4], bo[4];
    be[0] = *(const v2f*)(wb0); be[1] = *(const v2f*)(wb1);
    be[2] = *(const v2f*)(wb2); be[3] = *(const v2f*)(wb3);
#pragma unroll
    for (int st = 0; st < 8; st += 2) {
      bo[0] = *(const v2f*)(wb0 + 4 * (st + 1));
      bo[1] = *(const v2f*)(wb1 + 4 * (st + 1));
      bo[2] = *(const v2f*)(wb2 + 4 * (st + 1));
      bo[3] = *(const v2f*)(wb3 + 4 * (st + 1));
#pragma unroll
      for (int nt = 0; nt < 4; ++nt) {
        acc[0][nt] = wmma_f32(a0[st], be[nt], acc[0][nt]);
        acc[1][nt] = wmma_f32(a1[st], be[nt], acc[1][nt]);
      }
      if (st + 2 < 8) {
        be[0] = *(const v2f*)(wb0 + 4 * (st + 2));
        be[1] = *(const v2f*)(wb1 + 4 * (st + 2));
        be[2] = *(const v2f*)(wb2 + 4 * (st + 2));
        be[3] = *(const v2f*)(wb3 + 4 * (st + 2));
      }
#pragma unroll
      for (int nt = 0; nt < 4; ++nt) {
        acc[0][nt] = wmma_f32(a0[st + 1], bo[nt], acc[0][nt]);
        acc[1][nt] = wmma_f32(a1[st + 1], bo[nt], acc[1][nt]);
      }
    }

    // publish slab kb+1 (single barrier per slab)
#if HAVE_TDM
    if (wv == 0 && more) __builtin_amdgcn_s_wait_tensorcnt(0);
#else
    if (more) {
      float* wdst = &Wl[(kb + 1) & 1][sn * LDS_STRIDE + sc];
      *(v4f*)(wdst)     = st0;
      *(v4f*)(wdst + 4) = st1;
    }
#endif
    __syncthreads();
  }

#pragma unroll
  for (int mi = 0; mi < 2; ++mi)
#pragma unroll
    for (int nt = 0; nt < 4; ++nt)
#pragma unroll
      for (int i = 0; i < 8; ++i)
        Y[(size_t)(m0 + mi * 16 + i + 8 * hf) * N + n0 + nt * 16 + lm] =
            acc[mi][nt][i];
}

// One wave per (b, h, 16-row query tile). Two passes over causal key tiles:
// pass 1: row max + sum of exp.  pass 2: exact probs -> global, O += P @ V.
__global__ __launch_bounds__(128) void attn_kernel(const float* __restrict__ Q,
                                                   const float* __restrict__ K,
                                                   const float* __restrict__ V,
                                                   float* __restrict__ probs,
                                                   float* __restrict__ outp) {
  __shared__ __align__(16) float ptile[4][256];  // per-wave 16x16 P tile

  const int lane = threadIdx.x & 31;
  const int wv   = threadIdx.x >> 5;
  const int lm   = lane & 15;
  const int hf   = lane >> 4;

  int fw = blockIdx.x * 4 + wv;
  const int qt = fw & 127;            // query tile index (T/16 = 128)
  fw >>= 7;
  const int h = fw & 15;
  const int b = fw >> 4;
  const int q0 = qt * 16;

  const size_t headoff = (size_t)b * T_ * D_ + (size_t)h * HD_;
  const float* Qh = Q + headoff;
  const float* Kh = K + headoff;
  const float* Vh = V + headoff;

  // Q tile as WMMA A-fragments (K=64 -> 16 steps of 4); temperature folded in.
  v2f qa[16];
#pragma unroll
  for (int s = 0; s < 16; ++s) {
    v2f t = *(const v2f*)(Qh + (size_t)(q0 + lm) * D_ + 4 * s + 2 * hf);
    qa[s].x = t.x * INV_TEMP;
    qa[s].y = t.y * INV_TEMP;
  }

  // ---- pass 1: row max / row sum(exp) over causal keys ----
  float rm[8], rs[8];
#pragma unroll
  for (int i = 0; i < 8; ++i) { rm[i] = -INFINITY; rs[i] = 0.0f; }

  for (int j = 0; j <= qt; ++j) {
    const float* Kt = Kh + (size_t)(j * 16 + lm) * D_ + 2 * hf;
    v2f kb[16];
#pragma unroll
    for (int st = 0; st < 16; ++st) kb[st] = *(const v2f*)(Kt + 4 * st);
    v8f s = {};
#pragma unroll
    for (int st = 0; st < 16; ++st) s = wmma_f32(qa[st], kb[st], s);
#pragma unroll
    for (int i = 0; i < 8; ++i) {  // row m = i + 8*hf, key n = lm
      float v = s[i];
      if (j == qt && lm > i + 8 * hf) v = -INFINITY;
      float nm = fmaxf(rm[i], half_reduce_max(v));
      float ts = half_reduce_sum(__expf(v - nm));
      rs[i] = rs[i] * __expf(rm[i] - nm) + ts;
      rm[i] = nm;
    }
  }

  float il[8];
#pragma unroll
  for (int i = 0; i < 8; ++i) il[i] = 1.0f / rs[i];

  // ---- pass 2: exact probabilities + O accumulation ----
  v8f o0 = {}, o1 = {}, o2 = {}, o3 = {};
  float* arow = probs + ((size_t)(b * H_ + h) * T_ + q0) * T_;

  for (int j = 0; j <= qt; ++j) {
    const float* Kt = Kh + (size_t)(j * 16 + lm) * D_ + 2 * hf;
    v2f kb[16];
#pragma unroll
    for (int st = 0; st < 16; ++st) kb[st] = *(const v2f*)(Kt + 4 * st);
    v8f s = {};
#pragma unroll
    for (int st = 0; st < 16; ++st) s = wmma_f32(qa[st], kb[st], s);

#pragma unroll
    for (int i = 0; i < 8; ++i) {
      float v = s[i];
      if (j == qt && lm > i + 8 * hf) v = -INFINITY;
      float p = __expf(v - rm[i]) * il[i];
      arow[(size_t)(i + 8 * hf) * T_ + j * 16 + lm] = p;   // exact softmax out
      ptile[wv][(i + 8 * hf) * 16 + lm] = p;               // C-layout -> LDS
    }
    asm volatile("" ::: "memory");  // keep DS writes before DS reads

    // O += P(16x16) @ V(16x64): batch all fragment loads, then 16 WMMAs.
    v2f pa[4];
#pragma unroll
    for (int st = 0; st < 4; ++st)
      pa[st] = *(const v2f*)(&ptile[wv][lm * 16 + 4 * st + 2 * hf]);
    v2f vb[4][4];  // [st][nt]
#pragma unroll
    for (int st = 0; st < 4; ++st) {
      const float* Vt = Vh + (size_t)(j * 16 + 4 * st + 2 * hf) * D_;
#pragma unroll
      for (int nt = 0; nt < 4; ++nt) {
        vb[st][nt].x = Vt[nt * 16 + lm];
        vb[st][nt].y = Vt[D_ + nt * 16 + lm];
      }
    }
#pragma unroll
    for (int st = 0; st < 4; ++st) {
      o0 = wmma_f32(pa[st], vb[st][0], o0);
      o1 = wmma_f32(pa[st], vb[st][1], o1);
      o2 = wmma_f32(pa[st], vb[st][2], o2);
      o3 = wmma_f32(pa[st], vb[st][3], o3);
    }
    asm volatile("" ::: "memory");
  }

  // zero the masked upper triangle (poisoned buffer; ref softmax gives 0).
  // b128 stores: 16 lanes of a half cover 64 cols per step of one row.
  {
    const int c0 = (qt + 1) * 16;
    const v4f z = {0.0f, 0.0f, 0.0f, 0.0f};
#pragma unroll
    for (int i = 0; i < 8; ++i) {
      float* rp = arow + (size_t)(i + 8 * hf) * T_;
      for (int c = c0 + lm * 4; c < T_; c += 64) *(v4f*)(rp + c) = z;
    }
  }

  // merged-heads layout [B*T, D] at column h*64
  float* orow = outp + (size_t)(b * T_ + q0) * D_ + h * HD_;
#pragma unroll
  for (int i = 0; i < 8; ++i) {
    const size_t r = (size_t)(i + 8 * hf) * D_;
    orow[r + lm]      = o0[i];
    orow[r + 16 + lm] = o1[i];
    orow[r + 32 + lm] = o2[i];
    orow[r + 48 + lm] = o3[i];
  }
}

extern "C" void kernel_launch(void* const* d_in, const int* in_sizes, int n_in,
                              void* d_out, int out_size, void* d_ws,
                              size_t ws_size, hipStream_t stream) {
  const float* q  = (const float*)d_in[0];
  const float* k  = (const float*)d_in[1];
  const float* v  = (const float*)d_in[2];
  const float* Wq = (const float*)d_in[3];
  const float* Wk = (const float*)d_in[4];
  const float* Wv = (const float*)d_in[5];
  const float* Wc = (const float*)d_in[6];
  // d_in[7] = causal mask (bool) -- handled analytically, not read.

  const size_t SZ = (size_t)B_ * T_ * D_;
  float* out  = (float*)d_out;        // [B,T,D]
  float* attn = out + SZ;             // [B,H,T,T]
  float* ws   = (float*)d_ws;
  float* Qp = ws;
  float* Kp = ws + SZ;
  float* Vp = ws + 2 * SZ;
  float* Op = ws + 3 * SZ;

  const dim3 gg(D_ / 64, (B_ * T_) / 256);
  gemm_xwt<<<gg, 256, 0, stream>>>(q, Wq, Qp, B_ * T_, D_, D_);
  gemm_xwt<<<gg, 256, 0, stream>>>(k, Wk, Kp, B_ * T_, D_, D_);
  gemm_xwt<<<gg, 256, 0, stream>>>(v, Wv, Vp, B_ * T_, D_, D_);

  attn_kernel<<<(B_ * H_ * (T_ / 16)) / 4, 128, 0, stream>>>(Qp, Kp, Vp, attn,
                                                             Op);

  gemm_xwt<<<gg, 256, 0, stream>>>(Op, Wc, out, B_ * T_, D_, D_);
}